// Attention_RPEHP_28527172780790
// MI455X (gfx1250) — compile-verified
//
#include <hip/hip_runtime.h>
#include <hip/hip_bf16.h>

typedef float v2f __attribute__((ext_vector_type(2)));
typedef float v8f __attribute__((ext_vector_type(8)));

#define BATCH  8
#define HEADS  8
#define NQ     1024
#define DIM    512
#define DH     64
#define INNER  512
#define TRIPLE 1536

static __device__ __forceinline__ v8f wmma_f32_16x16x4(v2f a, v2f b, v8f c) {
  // D = A(16x4, f32) * B(4x16, f32) + C(16x16, f32)
  return __builtin_amdgcn_wmma_f32_16x16x4_f32(false, a, false, b, (short)0, c,
                                               false, false);
}

static __device__ __forceinline__ v8f v8f_zero() {
  v8f z = {0.f, 0.f, 0.f, 0.f, 0.f, 0.f, 0.f, 0.f};
  return z;
}

// ---------------------------------------------------------------------------
// Kernel 1: qkv = x @ w_qkv, scattered into qkv_cat [b, h, n, 3*DH]
// grid (12, 64, 8), block 256 (8 waves); wave owns one 16x16 output tile.
// ---------------------------------------------------------------------------
__global__ __launch_bounds__(256) void qkv_gemm_kernel(
    const float* __restrict__ x, const float* __restrict__ w,
    float* __restrict__ qkvcat) {
  __shared__ float sA[16 * DIM];  // 32 KB
  const int b = blockIdx.z;
  const int rt = blockIdx.y;
  const int ct = blockIdx.x;
  const int tid = threadIdx.x;
  const int wave = tid >> 5;
  const int lane = tid & 31;
  const int g = lane >> 4;   // half-wave id
  const int ln = lane & 15;  // lane within half

  const float* xb = x + ((size_t)b * NQ + (size_t)rt * 16) * DIM;
  for (int i = tid; i < 16 * DIM; i += 256) sA[i] = xb[i];  // rows contiguous
  __syncthreads();

  const int colbase = ct * 128 + wave * 16;
  v8f c = v8f_zero();
  for (int k = 0; k < DIM; k += 4) {
    const int kk = k + 2 * g;
    v2f a, bf;
    a.x = sA[ln * DIM + kk];
    a.y = sA[ln * DIM + kk + 1];
    bf.x = w[(size_t)kk * TRIPLE + colbase + ln];
    bf.y = w[(size_t)(kk + 1) * TRIPLE + colbase + ln];
    c = wmma_f32_16x16x4(a, bf, c);
  }

  // scatter into qkv_cat: col cc in [0,1536): p = cc/512, head, dd
  const int cc = colbase + ln;
  const int p = cc >> 9;
  const int rem = cc & 511;
  const int head = rem >> 6;
  const int dd = rem & 63;
#pragma unroll
  for (int r = 0; r < 8; ++r) {
    const int row = rt * 16 + r + 8 * g;
    const size_t dst =
        ((size_t)(b * HEADS + head) * NQ + row) * (3 * DH) + p * DH + dd;
    qkvcat[dst] = c[r];
  }
}

// ---------------------------------------------------------------------------
// Kernel 2: fused attention for one (b,h) and one 16-row q tile.
// grid (64 qtiles, 64 bh), block 256 (8 waves); wave owns 128 key columns.
// ---------------------------------------------------------------------------
__global__ __launch_bounds__(256) void attn_kernel(
    const float* __restrict__ qkvcat, const float* __restrict__ rpb,
    const float* __restrict__ headsita, const float* __restrict__ dis,
    const int* __restrict__ rpi, float* __restrict__ probs0,
    float* __restrict__ attn_ws) {
  __shared__ float sQ[16 * DH];          // 4 KB
  __shared__ float sAttn[8][16 * 64];    // 32 KB, per-wave transpose staging
  __shared__ float sRed[8][16];          // cross-wave row partials
  __shared__ float sRowA[16];            // row max
  __shared__ float sRowB[16];            // row sum
  __shared__ float sOut[16 * DH];        // 4 KB block accumulation

  const int bh = blockIdx.y;
  const int b = bh >> 3;
  const int h = bh & 7;
  const int qt = blockIdx.x;
  const int tid = threadIdx.x;
  const int wave = tid >> 5;
  const int lane = tid & 31;
  const int g = lane >> 4;
  const int ln = lane & 15;

  const float sita = headsita[h];
  const float factor = 1.0f / (2.0f * sita * sita + 1e-10f);

  const float* qbase = qkvcat + (size_t)bh * NQ * 192;
  const float* kbase = qbase + DH;
  const float* vbase = qbase + 2 * DH;

  // stage Q tile (16 x 64)
  for (int i = tid; i < 16 * DH; i += 256) {
    const int r = i >> 6, cc = i & 63;
    sQ[i] = qbase[(size_t)(qt * 16 + r) * 192 + cc];
  }
  __syncthreads();

  // ---- dots0 = scale * Q K^T : wave covers keys [wave*128, wave*128+128) ---
  v8f acc[8];
#pragma unroll
  for (int t = 0; t < 8; ++t) {
    v8f c = v8f_zero();
    const int colb = wave * 128 + t * 16;
    for (int k = 0; k < DH; k += 4) {
      const int kk = k + 2 * g;
      v2f a, bf;
      a.x = sQ[ln * DH + kk];
      a.y = sQ[ln * DH + kk + 1];
      bf.x = kbase[(size_t)(colb + ln) * 192 + kk];
      bf.y = kbase[(size_t)(colb + ln) * 192 + kk + 1];
      c = wmma_f32_16x16x4(a, bf, c);
    }
#pragma unroll
    for (int r = 0; r < 8; ++r) c[r] *= 0.125f;  // DIM_HEAD^-0.5
    acc[t] = c;
  }

  // ---- softmax stats for dots0 ----
  float rm[8], rinv[8];
#pragma unroll
  for (int r = 0; r < 8; ++r) {
    float m = acc[0][r];
#pragma unroll
    for (int t = 1; t < 8; ++t) m = fmaxf(m, acc[t][r]);
    for (int off = 8; off >= 1; off >>= 1) m = fmaxf(m, __shfl_xor(m, off));
    rm[r] = m;  // per-half row partial max
  }
  if (ln == 0)
#pragma unroll
    for (int r = 0; r < 8; ++r) sRed[wave][r + 8 * g] = rm[r];
  __syncthreads();
  if (tid < 16) {
    float m = sRed[0][tid];
    for (int w2 = 1; w2 < 8; ++w2) m = fmaxf(m, sRed[w2][tid]);
    sRowA[tid] = m;
  }
  __syncthreads();
#pragma unroll
  for (int r = 0; r < 8; ++r) rm[r] = sRowA[r + 8 * g];

#pragma unroll
  for (int r = 0; r < 8; ++r) {
    float s = 0.f;
#pragma unroll
    for (int t = 0; t < 8; ++t) s += __expf(acc[t][r] - rm[r]);
    for (int off = 8; off >= 1; off >>= 1) s += __shfl_xor(s, off);
    rinv[r] = s;
  }
  if (ln == 0)
#pragma unroll
    for (int r = 0; r < 8; ++r) sRed[wave][r + 8 * g] = rinv[r];
  __syncthreads();
  if (tid < 16) {
    float s = 0.f;
    for (int w2 = 0; w2 < 8; ++w2) s += sRed[w2][tid];
    sRowB[tid] = s;
  }
  __syncthreads();
#pragma unroll
  for (int r = 0; r < 8; ++r) rinv[r] = 1.0f / sRowB[r + 8 * g];

  // ---- write softmax(dots0) ----
  float* p0 = probs0 + ((size_t)bh * NQ + qt * 16) * NQ;
#pragma unroll
  for (int t = 0; t < 8; ++t) {
    const int colb = wave * 128 + t * 16;
#pragma unroll
    for (int r = 0; r < 8; ++r) {
      const int row = r + 8 * g;
      p0[(size_t)row * NQ + colb + ln] = __expf(acc[t][r] - rm[r]) * rinv[r];
    }
  }

  // ---- add rel-pos bias + Gaussian decay (gathered on the fly) ----
#pragma unroll
  for (int t = 0; t < 8; ++t) {
    const int colb = wave * 128 + t * 16;
#pragma unroll
    for (int r = 0; r < 8; ++r) {
      const int row = qt * 16 + r + 8 * g;
      const size_t ij = (size_t)row * NQ + colb + ln;
      const float bias =
          rpb[rpi[ij] * HEADS + h] + 0.01f * __expf(-factor * dis[ij]);
      acc[t][r] += bias;
    }
  }

  // ---- softmax stats for biased dots ----
  __syncthreads();
#pragma unroll
  for (int r = 0; r < 8; ++r) {
    float m = acc[0][r];
#pragma unroll
    for (int t = 1; t < 8; ++t) m = fmaxf(m, acc[t][r]);
    for (int off = 8; off >= 1; off >>= 1) m = fmaxf(m, __shfl_xor(m, off));
    rm[r] = m;
  }
  if (ln == 0)
#pragma unroll
    for (int r = 0; r < 8; ++r) sRed[wave][r + 8 * g] = rm[r];
  __syncthreads();
  if (tid < 16) {
    float m = sRed[0][tid];
    for (int w2 = 1; w2 < 8; ++w2) m = fmaxf(m, sRed[w2][tid]);
    sRowA[tid] = m;
  }
  __syncthreads();
#pragma unroll
  for (int r = 0; r < 8; ++r) rm[r] = sRowA[r + 8 * g];

#pragma unroll
  for (int r = 0; r < 8; ++r) {
    float s = 0.f;
#pragma unroll
    for (int t = 0; t < 8; ++t) s += __expf(acc[t][r] - rm[r]);
    for (int off = 8; off >= 1; off >>= 1) s += __shfl_xor(s, off);
    rinv[r] = s;
  }
  __syncthreads();
  if (ln == 0)
#pragma unroll
    for (int r = 0; r < 8; ++r) sRed[wave][r + 8 * g] = rinv[r];
  __syncthreads();
  if (tid < 16) {
    float s = 0.f;
    for (int w2 = 0; w2 < 8; ++w2) s += sRed[w2][tid];
    sRowB[tid] = s;
  }
  __syncthreads();
#pragma unroll
  for (int r = 0; r < 8; ++r) rinv[r] = 1.0f / sRowB[r + 8 * g];

  // ---- attn @ V : transpose probs through per-wave LDS, WMMA, reduce ----
  for (int i = tid; i < 16 * DH; i += 256) sOut[i] = 0.f;
  v8f oacc[4];
#pragma unroll
  for (int dt = 0; dt < 4; ++dt) oacc[dt] = v8f_zero();
  __syncthreads();

  for (int half = 0; half < 2; ++half) {
#pragma unroll
    for (int t = 0; t < 4; ++t) {
      const int tt = half * 4 + t;
#pragma unroll
      for (int r = 0; r < 8; ++r) {
        const int m = r + 8 * g;
        sAttn[wave][m * 64 + t * 16 + ln] =
            __expf(acc[tt][r] - rm[r]) * rinv[r];
      }
    }
    __syncthreads();
    const int keyb = wave * 128 + half * 64;
#pragma unroll
    for (int dt = 0; dt < 4; ++dt) {
      for (int k = 0; k < 64; k += 4) {
        const int kk = k + 2 * g;
        v2f a, bf;
        a.x = sAttn[wave][ln * 64 + kk];
        a.y = sAttn[wave][ln * 64 + kk + 1];
        bf.x = vbase[(size_t)(keyb + kk) * 192 + dt * 16 + ln];
        bf.y = vbase[(size_t)(keyb + kk + 1) * 192 + dt * 16 + ln];
        oacc[dt] = wmma_f32_16x16x4(a, bf, oacc[dt]);
      }
    }
    __syncthreads();
  }

#pragma unroll
  for (int dt = 0; dt < 4; ++dt)
#pragma unroll
    for (int r = 0; r < 8; ++r)
      atomicAdd(&sOut[(r + 8 * g) * 64 + dt * 16 + ln], oacc[dt][r]);
  __syncthreads();

  // store attn output in [b, n, h*64+dd] layout for the output projection
  float* ob = attn_ws + ((size_t)b * NQ + qt * 16) * INNER + h * DH;
  for (int i = tid; i < 16 * DH; i += 256) {
    const int r = i >> 6, dd = i & 63;
    ob[(size_t)r * INNER + dd] = sOut[i];
  }
}

// ---------------------------------------------------------------------------
// Kernel 3: out = attn_ws @ w_out + b_out
// grid (4, 64, 8), block 256 (8 waves).
// ---------------------------------------------------------------------------
__global__ __launch_bounds__(256) void out_gemm_kernel(
    const float* __restrict__ attn_ws, const float* __restrict__ w,
    const float* __restrict__ bias, float* __restrict__ out) {
  __shared__ float sA[16 * INNER];  // 32 KB
  const int b = blockIdx.z;
  const int rt = blockIdx.y;
  const int ct = blockIdx.x;
  const int tid = threadIdx.x;
  const int wave = tid >> 5;
  const int lane = tid & 31;
  const int g = lane >> 4;
  const int ln = lane & 15;

  const float* ab = attn_ws + ((size_t)b * NQ + (size_t)rt * 16) * INNER;
  for (int i = tid; i < 16 * INNER; i += 256) sA[i] = ab[i];
  __syncthreads();

  const int colbase = ct * 128 + wave * 16;
  v8f c = v8f_zero();
  for (int k = 0; k < INNER; k += 4) {
    const int kk = k + 2 * g;
    v2f a, bf;
    a.x = sA[ln * INNER + kk];
    a.y = sA[ln * INNER + kk + 1];
    bf.x = w[(size_t)kk * DIM + colbase + ln];
    bf.y = w[(size_t)(kk + 1) * DIM + colbase + ln];
    c = wmma_f32_16x16x4(a, bf, c);
  }
  const float bo = bias[colbase + ln];
#pragma unroll
  for (int r = 0; r < 8; ++r) {
    const int row = rt * 16 + r + 8 * g;
    out[((size_t)b * NQ + row) * DIM + colbase + ln] = c[r] + bo;
  }
}

// ---------------------------------------------------------------------------
extern "C" void kernel_launch(void* const* d_in, const int* in_sizes, int n_in,
                              void* d_out, int out_size, void* d_ws,
                              size_t ws_size, hipStream_t stream) {
  const float* x        = (const float*)d_in[0];
  const float* w_qkv    = (const float*)d_in[1];
  const float* rpb      = (const float*)d_in[2];
  const float* headsita = (const float*)d_in[3];
  const float* w_out    = (const float*)d_in[4];
  const float* b_out    = (const float*)d_in[5];
  const float* dis      = (const float*)d_in[6];
  const int*   rpi      = (const int*)d_in[7];

  float* out    = (float*)d_out;                                   // [8,1024,512]
  float* qkvcat = out + (size_t)BATCH * NQ * DIM;                  // [8,8,1024,192]
  float* probs0 = qkvcat + (size_t)BATCH * HEADS * NQ * 3 * DH;    // [8,8,1024,1024]
  float* attn_ws = (float*)d_ws;                                   // [8,1024,512]

  qkv_gemm_kernel<<<dim3(12, 64, 8), 256, 0, stream>>>(x, w_qkv, qkvcat);
  attn_kernel<<<dim3(64, 64), 256, 0, stream>>>(qkvcat, rpb, headsita, dis,
                                                rpi, probs0, attn_ws);
  out_gemm_kernel<<<dim3(4, 64, 8), 256, 0, stream>>>(attn_ws, w_out, b_out,
                                                      out);
}